// Polarization_57131654971473
// MI455X (gfx1250) — compile-verified
//
#include <hip/hip_runtime.h>
#include <math.h>

// ---------------------------------------------------------------------------
// Thole mutual polarization on MI455X (gfx1250, wave32).
// Memory-bound sparse CG solve; tij precomputed once (SoA, 6 floats/edge).
// Dot-product block reductions use V_WMMA_F32_16X16X4_F32 (ones-matrix trick).
// ---------------------------------------------------------------------------

#define BOHR_F 0.52917721067f
#define DAMP_F 0.39f

typedef __attribute__((ext_vector_type(2))) float v2f;
typedef __attribute__((ext_vector_type(8))) float v8f;

enum { S_RSOLD = 0, S_PAP = 1, S_RSNEW = 2, S_ALPHA = 3, S_BETA = 4, S_COUNT = 8 };

// Reduce 256 per-thread partials to one float and atomically add to *target.
// Wave 0 (lanes 0..31, EXEC all-ones) does 4 chained f32 WMMAs:
//   D = A(16x4 partials) x B(4x16 ones) + C  => D[m][n] = rowsum_m
// Lane l<16 holds S_0..S_7 in its 8 D VGPRs, lane l>=16 holds S_8..S_15,
// so per-lane sum + shfl_xor(16) yields the full 256-element sum.
__device__ __forceinline__ void block_reduce_add_256(float v, float* target) {
    __shared__ float sdata[256];
    const int tid = threadIdx.x;
    sdata[tid] = v;
    __syncthreads();
    if (tid < 32) {  // whole wave 0 -> EXEC all ones inside (WMMA requirement)
        const int m  = tid & 15;
        const int kb = (tid >> 4) << 1;  // lanes 0-15: K'={0,1}; lanes 16-31: K'={2,3}
        v2f ones; ones.x = 1.0f; ones.y = 1.0f;
        v8f acc = {0.f, 0.f, 0.f, 0.f, 0.f, 0.f, 0.f, 0.f};
#pragma unroll
        for (int c = 0; c < 4; ++c) {
            v2f a;
            a.x = sdata[m * 16 + c * 4 + kb];
            a.y = sdata[m * 16 + c * 4 + kb + 1];
            acc = __builtin_amdgcn_wmma_f32_16x16x4_f32(
                false, a, false, ones, (short)0, acc, false, false);
        }
        float s = acc[0] + acc[1] + acc[2] + acc[3] + acc[4] + acc[5] + acc[6] + acc[7];
        s += __shfl_xor(s, 16);
        if (tid == 0) atomicAdd(target, s);
    }
}

// --- one-time precompute ----------------------------------------------------

// Per-edge symmetric damped dipole tensor, SoA: t[q*E + e], q in
// {xx,yy,zz,xy,xz,yz}. Coalesced b32 streams in the CG hot loop.
__global__ void pol_precompute_edges(int E,
                                     const int* __restrict__ src,
                                     const int* __restrict__ dst,
                                     const float* __restrict__ dist,
                                     const float* __restrict__ vec,
                                     const float* __restrict__ pol,
                                     float* __restrict__ t) {
    int e = blockIdx.x * blockDim.x + threadIdx.x;
    if (e >= E) return;
    const float invb  = 1.0f / BOHR_F;
    const float invb3 = invb * invb * invb;
    float rij = dist[e] * invb;
    float vx  = vec[3 * e + 0] * invb;
    float vy  = vec[3 * e + 1] * invb;
    float vz  = vec[3 * e + 2] * invb;
    float ps  = pol[src[e]] * invb3;
    float pd  = pol[dst[e]] * invb3;
    float aij = pd * ps;
    float u   = rij * powf(aij, -1.0f / 6.0f);
    float u3  = DAMP_F * u * u * u;
    float ex  = expf(-u3);
    float l3  = 1.0f - ex;
    float l5  = 1.0f - (1.0f + u3) * ex;
    float r2     = rij * rij;
    float inv_r3 = 1.0f / (r2 * rij);
    float inv_r5 = inv_r3 / r2;
    float c5 = 3.0f * l5 * inv_r5;
    float c3 = l3 * inv_r3;
    t[0 * E + e] = c5 * vx * vx - c3;
    t[1 * E + e] = c5 * vy * vy - c3;
    t[2 * E + e] = c5 * vz * vz - c3;
    t[3 * E + e] = c5 * vx * vy;
    t[4 * E + e] = c5 * vx * vz;
    t[5 * E + e] = c5 * vy * vz;
}

__global__ void pol_diag_kernel(int N, const float* __restrict__ pol,
                                float* __restrict__ diag) {
    int i = blockIdx.x * blockDim.x + threadIdx.x;
    if (i >= N) return;
    diag[i] = (BOHR_F * BOHR_F * BOHR_F) / pol[i];  // 1 / (pol/BOHR^3)
}

__global__ void pol_zero_scalars(float* __restrict__ sc) {
    if (threadIdx.x < S_COUNT) sc[threadIdx.x] = 0.0f;
}

// x=0, r=b, p=b, rsold = b.b (block-reduced via WMMA)
__global__ void pol_cg_init(int n3, const float* __restrict__ b,
                            float* __restrict__ mu, float* __restrict__ r,
                            float* __restrict__ p, float* __restrict__ sc) {
    int k = blockIdx.x * blockDim.x + threadIdx.x;
    float v = 0.0f;
    if (k < n3) {
        float bk = b[k];
        mu[k] = 0.0f;
        r[k]  = bk;
        p[k]  = bk;
        v = bk * bk;
    }
    block_reduce_add_256(v, &sc[S_RSOLD]);
}

// --- matvec: Ap = diag*p  +  scatter_src( T_ij * p[dst] ) ------------------

__global__ void pol_matvec_init(int n3, const float* __restrict__ pin,
                                const float* __restrict__ diag,
                                float* __restrict__ Ap, float* __restrict__ sc) {
    int k = blockIdx.x * blockDim.x + threadIdx.x;
    if (k == 0) sc[S_PAP] = 0.0f;  // safe: dot kernel launches after this one
    if (k >= n3) return;
    Ap[k] = diag[k / 3] * pin[k];
}

__global__ void pol_matvec_edges(int E,
                                 const int* __restrict__ src,
                                 const int* __restrict__ dst,
                                 const float* __restrict__ t,
                                 const float* __restrict__ pin,
                                 float* __restrict__ Ap) {
    int e = blockIdx.x * blockDim.x + threadIdx.x;
    if (e >= E) return;
    int j = dst[e];
    float px = pin[3 * j + 0];  // gather: p-vector is 1.2 MB -> L2-resident
    float py = pin[3 * j + 1];
    float pz = pin[3 * j + 2];
    float xx = t[0 * E + e], yy = t[1 * E + e], zz = t[2 * E + e];
    float xy = t[3 * E + e], xz = t[4 * E + e], yz = t[5 * E + e];
    float ox = xx * px + xy * py + xz * pz;
    float oy = xy * px + yy * py + yz * pz;
    float oz = xz * px + yz * py + zz * pz;
    int i = src[e];
    atomicAdd(&Ap[3 * i + 0], ox);
    atomicAdd(&Ap[3 * i + 1], oy);
    atomicAdd(&Ap[3 * i + 2], oz);
}

// --- CG update kernels ------------------------------------------------------

__global__ void pol_dot_kernel(int n3, const float* __restrict__ a,
                               const float* __restrict__ b,
                               float* __restrict__ target) {
    int k = blockIdx.x * blockDim.x + threadIdx.x;
    float v = (k < n3) ? a[k] * b[k] : 0.0f;
    block_reduce_add_256(v, target);
}

__global__ void pol_scalar_alpha(float* __restrict__ sc) {
    sc[S_ALPHA] = sc[S_RSOLD] / sc[S_PAP];
    sc[S_RSNEW] = 0.0f;
}

__global__ void pol_axpy_rr(int n3, const float* __restrict__ sc_in,
                            const float* __restrict__ p, const float* __restrict__ Ap,
                            float* __restrict__ mu, float* __restrict__ r,
                            float* __restrict__ sc_out) {
    int k = blockIdx.x * blockDim.x + threadIdx.x;
    float alpha = sc_in[S_ALPHA];
    float v = 0.0f;
    if (k < n3) {
        mu[k] += alpha * p[k];
        float rk = r[k] - alpha * Ap[k];
        r[k] = rk;
        v = rk * rk;
    }
    block_reduce_add_256(v, &sc_out[S_RSNEW]);
}

__global__ void pol_scalar_beta(float* __restrict__ sc) {
    float rs = sc[S_RSNEW];
    sc[S_BETA]  = rs / sc[S_RSOLD];
    sc[S_RSOLD] = rs;
}

__global__ void pol_p_update(int n3, const float* __restrict__ sc,
                             const float* __restrict__ r, float* __restrict__ p) {
    int k = blockIdx.x * blockDim.x + threadIdx.x;
    if (k >= n3) return;
    p[k] = r[k] + sc[S_BETA] * p[k];
}

// --- final energy -----------------------------------------------------------

__global__ void pol_energy_kernel(int N, const float* __restrict__ mu,
                                  const float* __restrict__ tmu,
                                  const float* __restrict__ ef,
                                  float* __restrict__ out) {
    int i = blockIdx.x * blockDim.x + threadIdx.x;
    if (i >= N) return;
    float acc = 0.0f;
#pragma unroll
    for (int a = 0; a < 3; ++a) {
        int k = 3 * i + a;
        acc += (0.5f * tmu[k] - ef[k]) * mu[k];
    }
    out[i] = acc;
}

// ---------------------------------------------------------------------------

extern "C" void kernel_launch(void* const* d_in, const int* in_sizes, int n_in,
                              void* d_out, int out_size, void* d_ws, size_t ws_size,
                              hipStream_t stream) {
    (void)n_in; (void)out_size; (void)ws_size;
    const int*   edge_src  = (const int*)d_in[1];
    const int*   edge_dst  = (const int*)d_in[2];
    const float* distances = (const float*)d_in[3];
    const float* vec       = (const float*)d_in[4];
    const float* pol       = (const float*)d_in[5];
    const float* efield    = (const float*)d_in[6];
    float* out = (float*)d_out;

    const int N  = in_sizes[0];   // species count == n_atoms
    const int E  = in_sizes[1];
    const int n3 = 3 * N;

    // Workspace layout (floats): tij SoA (6E) | diag (N) | mu,r,p,Ap (4*3N) | scalars (8)
    float* w    = (float*)d_ws;
    float* t    = w;
    float* diag = t + (size_t)6 * E;
    float* mu   = diag + N;
    float* rv   = mu + n3;
    float* pv   = rv + n3;
    float* Ap   = pv + n3;
    float* sc   = Ap + n3;

    const int TB = 256;                    // 8 waves/block on wave32
    const int gE = (E + TB - 1) / TB;
    const int g3 = (n3 + TB - 1) / TB;
    const int gN = (N + TB - 1) / TB;

    pol_precompute_edges<<<gE, TB, 0, stream>>>(E, edge_src, edge_dst,
                                                distances, vec, pol, t);
    pol_diag_kernel<<<gN, TB, 0, stream>>>(N, pol, diag);
    pol_zero_scalars<<<1, 32, 0, stream>>>(sc);
    pol_cg_init<<<g3, TB, 0, stream>>>(n3, efield, mu, rv, pv, sc);

    const int NITER = 64;  // diag-dominant Thole system; converges well past 1e-5
    for (int it = 0; it < NITER; ++it) {
        pol_matvec_init<<<g3, TB, 0, stream>>>(n3, pv, diag, Ap, sc);
        pol_matvec_edges<<<gE, TB, 0, stream>>>(E, edge_src, edge_dst, t, pv, Ap);
        pol_dot_kernel<<<g3, TB, 0, stream>>>(n3, pv, Ap, sc + S_PAP);
        pol_scalar_alpha<<<1, 1, 0, stream>>>(sc);
        pol_axpy_rr<<<g3, TB, 0, stream>>>(n3, sc, pv, Ap, mu, rv, sc);
        pol_scalar_beta<<<1, 1, 0, stream>>>(sc);
        pol_p_update<<<g3, TB, 0, stream>>>(n3, sc, rv, pv);
    }

    // tmu = matvec(mu), then per-atom energy
    pol_matvec_init<<<g3, TB, 0, stream>>>(n3, mu, diag, Ap, sc);
    pol_matvec_edges<<<gE, TB, 0, stream>>>(E, edge_src, edge_dst, t, mu, Ap);
    pol_energy_kernel<<<gN, TB, 0, stream>>>(N, mu, Ap, efield, out);
}